// SAGPool_82884278878470
// MI455X (gfx1250) — compile-verified
//
#include <hip/hip_runtime.h>
#include <hip/hip_bf16.h>

// SAGPool forward for MI455X (gfx1250).
// Key algorithmic move: (adj @ xf) @ w  ==  adj @ (xf @ w)  (w is 768x1),
// collapsing a 51.5-GFLOP fp32 GEMM into two GEMVs; the whole problem becomes
// HBM-bandwidth bound (~335 MB -> ~14 us at 23.3 TB/s).
// The adj*s GEMV runs on V_WMMA_F32_16X16X4_F32 (exact f32 semantics).
// Sigmoid is monotone -> top-k on pre-activation h is identical; skipped.

typedef float v2f __attribute__((ext_vector_type(2)));
typedef float v8f __attribute__((ext_vector_type(8)));

#define B_  32
#define F_  64
#define N_  1024
#define T_  12
#define K_  (F_ * T_)   // 768
#define P_  (N_ / 2)    // 512

// ---------------------------------------------------------------------------
// Kernel 1: s[b][m] = sum_{t,f} x[b,f,m,t] * w[t*F + f]
// One thread per (b, m). Weight (3 KB) staged in LDS, x rows read as float4.
// ---------------------------------------------------------------------------
__global__ __launch_bounds__(256) void sagpool_s_kernel(
    const float* __restrict__ x, const float* __restrict__ w,
    float* __restrict__ s) {
  __shared__ float w_lds[K_];
  for (int i = threadIdx.x; i < K_; i += 256) w_lds[i] = w[i];
  __syncthreads();

  const int idx = blockIdx.x * 256 + threadIdx.x;   // b*N + m
  const int b = idx >> 10;
  const int m = idx & (N_ - 1);
  const float* xb = x + (size_t)b * F_ * N_ * T_ + (size_t)m * T_;

  float acc = 0.0f;
  for (int f = 0; f < F_; ++f) {
    const float4* xr = (const float4*)(xb + (size_t)f * (N_ * T_));
    float4 a0 = xr[0], a1 = xr[1], a2 = xr[2];   // 12 contiguous floats
    acc += a0.x * w_lds[0 * F_ + f] + a0.y * w_lds[1 * F_ + f]
         + a0.z * w_lds[2 * F_ + f] + a0.w * w_lds[3 * F_ + f]
         + a1.x * w_lds[4 * F_ + f] + a1.y * w_lds[5 * F_ + f]
         + a1.z * w_lds[6 * F_ + f] + a1.w * w_lds[7 * F_ + f]
         + a2.x * w_lds[8 * F_ + f] + a2.y * w_lds[9 * F_ + f]
         + a2.z * w_lds[10 * F_ + f] + a2.w * w_lds[11 * F_ + f];
  }
  s[idx] = acc;
}

// ---------------------------------------------------------------------------
// Kernel 2: h[b][n] = sum_m adj[b][n][m] * s[b][m]   via WMMA f32 16x16x4.
// Block = 256 threads (8 waves), handles one batch and 128 rows.
// Wave handles 16 rows. A-tile layout (ISA 7.12.2, 32-bit A 16x4):
//   lane L<16 : V0=adj[row][m0+0], V1=adj[row][m0+1]
//   lane L>=16: V0=adj[row][m0+2], V1=adj[row][m0+3]
// -> one b64 load at arow + m0 + 2*hi. B-tile (4x16, columns replicated with
// s) mirrors the same K placement: V0/V1 = s[m0+2*hi], s[m0+1+2*hi] from LDS.
// All columns of D are equal, so lane 0 holds rows n0..n0+7 in c[0..7] and
// lane 16 holds rows n0+8..n0+15.
// ---------------------------------------------------------------------------
__global__ __launch_bounds__(256) void sagpool_score_kernel(
    const float* __restrict__ adj, const float* __restrict__ s,
    float* __restrict__ h) {
  __shared__ float s_lds[N_];
  const int b     = blockIdx.x >> 3;          // 8 chunks per batch
  const int chunk = (blockIdx.x & 7) * 128;

  const float* sb = s + (size_t)b * N_;
  for (int i = threadIdx.x; i < N_; i += 256) s_lds[i] = sb[i];
  __syncthreads();

  const int wave = threadIdx.x >> 5;
  const int lane = threadIdx.x & 31;
  const int hi   = lane >> 4;                 // 0: K pair {0,1}; 1: {2,3}
  const int n0   = chunk + wave * 16;
  const int row  = n0 + (lane & 15);
  const float* arow = adj + ((size_t)b * N_ + row) * N_ + 2 * hi;
  const float* srow = s_lds + 2 * hi;

  v8f c0 = {}, c1 = {}, c2 = {}, c3 = {};
  for (int m0 = 0; m0 < N_; m0 += 16) {
    v2f a0 = *(const v2f*)(arow + m0 + 0);
    v2f a1 = *(const v2f*)(arow + m0 + 4);
    v2f a2 = *(const v2f*)(arow + m0 + 8);
    v2f a3 = *(const v2f*)(arow + m0 + 12);
    v2f b0 = *(const v2f*)(srow + m0 + 0);
    v2f b1 = *(const v2f*)(srow + m0 + 4);
    v2f b2 = *(const v2f*)(srow + m0 + 8);
    v2f b3 = *(const v2f*)(srow + m0 + 12);
    c0 = __builtin_amdgcn_wmma_f32_16x16x4_f32(false, a0, false, b0, (short)0, c0, false, false);
    c1 = __builtin_amdgcn_wmma_f32_16x16x4_f32(false, a1, false, b1, (short)0, c1, false, false);
    c2 = __builtin_amdgcn_wmma_f32_16x16x4_f32(false, a2, false, b2, (short)0, c2, false, false);
    c3 = __builtin_amdgcn_wmma_f32_16x16x4_f32(false, a3, false, b3, (short)0, c3, false, false);
  }
  v8f c = (c0 + c1) + (c2 + c3);

  if ((lane & 15) == 0) {                     // lanes 0 and 16
    float* out = h + (size_t)b * N_ + n0 + hi * 8;
#pragma unroll
    for (int r = 0; r < 8; ++r) out[r] = c[r];
  }
}

// ---------------------------------------------------------------------------
// Kernel 3: stable descending rank per batch (== jax.lax.top_k ordering).
// One 1024-thread block per batch; O(N^2) LDS-broadcast scan (trivial cost).
// Sigmoid omitted: strictly monotone, identical selection/order.
// ---------------------------------------------------------------------------
__global__ __launch_bounds__(1024) void sagpool_topk_kernel(
    const float* __restrict__ h, int* __restrict__ top_idx) {
  __shared__ float sc[N_];
  const int b = blockIdx.x;
  const int i = threadIdx.x;
  const float v = h[(size_t)b * N_ + i];
  sc[i] = v;
  __syncthreads();

  int rank = 0;
  for (int j = 0; j < N_; ++j) {
    const float u = sc[j];
    rank += (u > v) || ((u == v) && (j < i));
  }
  if (rank < P_) top_idx[b * P_ + rank] = i;
}

// ---------------------------------------------------------------------------
// Kernel 4: out[b,f,p,0:12] = x[b,f,top_idx[b,p],0:12]. One thread per
// (b,f,p); 3 float4 copies; writes fully coalesced.
// ---------------------------------------------------------------------------
__global__ __launch_bounds__(256) void sagpool_gather_kernel(
    const float* __restrict__ x, const int* __restrict__ top_idx,
    float* __restrict__ out) {
  const int idx = blockIdx.x * 256 + threadIdx.x;   // (b*F + f)*P + p
  const int p  = idx & (P_ - 1);
  const int bf = idx >> 9;
  const int b  = bf >> 6;
  const int node = top_idx[b * P_ + p];
  const float4* src = (const float4*)(x + ((size_t)bf * N_ + node) * T_);
  float4*       dst = (float4*)(out + ((size_t)bf * P_ + p) * T_);
  dst[0] = src[0];
  dst[1] = src[1];
  dst[2] = src[2];
}

// ---------------------------------------------------------------------------
extern "C" void kernel_launch(void* const* d_in, const int* in_sizes, int n_in,
                              void* d_out, int out_size, void* d_ws, size_t ws_size,
                              hipStream_t stream) {
  const float* x   = (const float*)d_in[0];   // [32, 64, 1024, 12]
  const float* adj = (const float*)d_in[1];   // [32, 1024, 1024]
  const float* w   = (const float*)d_in[2];   // [768, 1]
  float* out = (float*)d_out;                 // [32, 64, 512, 12]

  // workspace layout: s (128 KB) | h (128 KB) | top_idx (64 KB)
  float* s       = (float*)d_ws;
  float* h       = s + (size_t)B_ * N_;
  int*   top_idx = (int*)(h + (size_t)B_ * N_);

  sagpool_s_kernel<<<(B_ * N_) / 256, 256, 0, stream>>>(x, w, s);
  sagpool_score_kernel<<<B_ * 8, 256, 0, stream>>>(adj, s, h);
  sagpool_topk_kernel<<<B_, 1024, 0, stream>>>(h, top_idx);
  sagpool_gather_kernel<<<(B_ * F_ * P_) / 256, 256, 0, stream>>>(x, top_idx, out);
}